// MolecularGNN_68779606278869
// MI455X (gfx1250) — compile-verified
//
#include <hip/hip_runtime.h>

#define NNODES 100000
#define NEDGES 1600000
#define FIN    64
#define HDIM   128
#define NL     64
#define NG     2048
#define BN_EPS 1e-5f

typedef __attribute__((ext_vector_type(2))) float v2f;
typedef __attribute__((ext_vector_type(8))) float v8f;

__device__ __forceinline__ void atomicAddF(float* p, float v) {
    __hip_atomic_fetch_add(p, v, __ATOMIC_RELAXED, __HIP_MEMORY_SCOPE_AGENT);
}

// ---------------------------------------------------------------------------
// fp32 WMMA GEMM: C[M x Nc] = A[M x K] @ B[K x Nc] (+bias, +relu optional).
// One wave32 per 16x16 output tile, V_WMMA_F32_16X16X4_F32, K stepped by 4.
// A layout per lane: m = lane%16, k0 = (lane/16)*2 -> {A[m][k0], A[m][k0+1]}
// B layout per lane: n = lane%16, k0 = (lane/16)*2 -> {B[k0][n], B[k0+1][n]}
// C layout: VGPR r -> row r (lanes 0-15) / row r+8 (lanes 16-31), col lane%16
// ---------------------------------------------------------------------------
__global__ void wmma_gemm_f32(const float* __restrict__ A,
                              const float* __restrict__ B,
                              float* __restrict__ C,
                              const float* __restrict__ bias,
                              int M, int K, int Nc, int relu) {
    const int tilesN = Nc >> 4;
    const int tile   = blockIdx.x;
    const int tm     = tile / tilesN;
    const int tn     = tile - tm * tilesN;
    const int lane   = threadIdx.x;          // 0..31
    const int r16    = lane & 15;
    const int khalf  = (lane >> 4) << 1;     // 0 or 2

    const float* arow = A + (size_t)(tm * 16 + r16) * K;
    const float* bcol = B + (size_t)(tn * 16 + r16);

    v8f c = {0.f, 0.f, 0.f, 0.f, 0.f, 0.f, 0.f, 0.f};
    for (int k = 0; k < K; k += 4) {
        v2f a, b;
        a.x = arow[k + khalf];
        a.y = arow[k + khalf + 1];
        b.x = bcol[(size_t)(k + khalf)     * Nc];
        b.y = bcol[(size_t)(k + khalf + 1) * Nc];
        c = __builtin_amdgcn_wmma_f32_16x16x4_f32(false, a, false, b,
                                                  (short)0, c, false, false);
    }

    const int col     = tn * 16 + r16;
    const int rowbase = tm * 16 + ((lane >> 4) << 3);  // +0 or +8
    const float bv    = bias ? bias[col] : 0.f;
#pragma unroll
    for (int i = 0; i < 8; ++i) {
        float v = c[i] + bv;
        if (relu) v = fmaxf(v, 0.f);
        C[(size_t)(rowbase + i) * Nc + col] = v;
    }
}

// ---------------------------------------------------------------------------
// degree accumulation over dst (reference: deg = segsum(1, dst) + 1)
// ---------------------------------------------------------------------------
__global__ void deg_kernel(const int* __restrict__ dst, float* __restrict__ deg) {
    int i = blockIdx.x * blockDim.x + threadIdx.x;
    if (i < NEDGES) atomicAddF(&deg[dst[i]], 1.0f);
}

__global__ void dinv_kernel(float* __restrict__ dinv) {
    int i = blockIdx.x * blockDim.x + threadIdx.x;
    if (i < NNODES) dinv[i] = rsqrtf(dinv[i] + 1.0f);
}

// ---------------------------------------------------------------------------
// edge aggregation: agg[dst] += hw[src] * dinv[src]*dinv[dst]
// one wave32 per edge: lane handles 4 consecutive floats (float4 load,
// 4 native f32 atomic adds) -> coalesced 512B row per edge.
// ---------------------------------------------------------------------------
__global__ void edge_agg(const float* __restrict__ hw,
                         const int* __restrict__ src,
                         const int* __restrict__ dst,
                         const float* __restrict__ dinv,
                         float* __restrict__ agg) {
    long long idx = (long long)blockIdx.x * blockDim.x + threadIdx.x;
    long long total = (long long)NEDGES * (HDIM / 4);
    if (idx >= total) return;
    int e  = (int)(idx >> 5);          // HDIM/4 == 32 lanes per edge
    int f4 = (int)(idx & 31) << 2;
    int s = src[e], d = dst[e];
    float nrm = dinv[s] * dinv[d];
    const float4 v = *(const float4*)(hw + (size_t)s * HDIM + f4);
    float* ap = agg + (size_t)d * HDIM + f4;
    atomicAddF(ap + 0, v.x * nrm);
    atomicAddF(ap + 1, v.y * nrm);
    atomicAddF(ap + 2, v.z * nrm);
    atomicAddF(ap + 3, v.w * nrm);
}

// ---------------------------------------------------------------------------
// fused: agg += hw*dinv^2 (self loop) + bias; BatchNorm(eval); ReLU  (in place)
// ---------------------------------------------------------------------------
__global__ void bn_relu_self(const float* __restrict__ hw,
                             const float* __restrict__ dinv,
                             const float* __restrict__ b,
                             const float* __restrict__ gamma,
                             const float* __restrict__ beta,
                             const float* __restrict__ mean,
                             const float* __restrict__ var,
                             float* __restrict__ agg) {
    long long idx = (long long)blockIdx.x * blockDim.x + threadIdx.x;
    if (idx >= (long long)NNODES * HDIM) return;
    int i = (int)(idx >> 7);          // / HDIM
    int c = (int)(idx & (HDIM - 1));
    float di = dinv[i];
    float v = agg[idx] + hw[idx] * di * di + b[c];
    v = (v - mean[c]) * rsqrtf(var[c] + BN_EPS) * gamma[c] + beta[c];
    agg[idx] = fmaxf(v, 0.f);
}

// ---------------------------------------------------------------------------
// global mean pool
// ---------------------------------------------------------------------------
__global__ void count_kernel(const int* __restrict__ batch, float* __restrict__ counts) {
    int i = blockIdx.x * blockDim.x + threadIdx.x;
    if (i < NNODES) atomicAddF(&counts[batch[i]], 1.0f);
}

__global__ void pool_sum(const float* __restrict__ h,
                         const int* __restrict__ batch,
                         float* __restrict__ pooled) {
    long long idx = (long long)blockIdx.x * blockDim.x + threadIdx.x;
    if (idx >= (long long)NNODES * HDIM) return;
    int i = (int)(idx >> 7);
    int c = (int)(idx & (HDIM - 1));
    atomicAddF(&pooled[(size_t)batch[i] * HDIM + c], h[idx]);
}

__global__ void pool_div(float* __restrict__ pooled, const float* __restrict__ counts) {
    int idx = blockIdx.x * blockDim.x + threadIdx.x;
    if (idx >= NG * HDIM) return;
    pooled[idx] /= fmaxf(counts[idx >> 7], 1.0f);
}

// ---------------------------------------------------------------------------
extern "C" void kernel_launch(void* const* d_in, const int* in_sizes, int n_in,
                              void* d_out, int out_size, void* d_ws, size_t ws_size,
                              hipStream_t stream) {
    const float* x      = (const float*)d_in[0];
    const int*   edge   = (const int*)d_in[1];
    const int*   src    = edge;
    const int*   dst    = edge + NEDGES;
    const int*   batch  = (const int*)d_in[2];
    const float* W[3]   = {(const float*)d_in[3], (const float*)d_in[5], (const float*)d_in[7]};
    const float* b[3]   = {(const float*)d_in[4], (const float*)d_in[6], (const float*)d_in[8]};
    const float* bn_g   = (const float*)d_in[9];
    const float* bn_b   = (const float*)d_in[10];
    const float* bn_m   = (const float*)d_in[11];
    const float* bn_v   = (const float*)d_in[12];
    const float* mlp_w1 = (const float*)d_in[13];
    const float* mlp_b1 = (const float*)d_in[14];
    const float* mlp_w2 = (const float*)d_in[15];
    const float* mlp_b2 = (const float*)d_in[16];

    // workspace layout (floats)
    float* ws = (float*)d_ws;
    size_t off = 0;
    float* dinv   = ws + off; off += 100352;                 // N padded
    float* hw     = ws + off; off += (size_t)NNODES * HDIM;  // h @ W
    float* bufA   = ws + off; off += (size_t)NNODES * HDIM;
    float* bufB   = ws + off; off += (size_t)NNODES * HDIM;
    float* pooled = ws + off; off += (size_t)NG * HDIM;
    float* counts = ws + off; off += NG;
    float* z1     = ws + off; off += (size_t)NG * HDIM;

    const int TB = 256;
    const long long nodeFeat = (long long)NNODES * HDIM;
    const int nfBlocks = (int)((nodeFeat + TB - 1) / TB);

    // degrees -> dinv
    hipMemsetAsync(dinv, 0, (size_t)NNODES * sizeof(float), stream);
    deg_kernel<<<(NEDGES + TB - 1) / TB, TB, 0, stream>>>(dst, dinv);
    dinv_kernel<<<(NNODES + TB - 1) / TB, TB, 0, stream>>>(dinv);

    // three GCN layers (ping-pong agg buffers, BN writes in place)
    const float* hin = x;
    float* aggbuf[3] = {bufA, bufB, bufA};
    int    Kdim[3]   = {FIN, HDIM, HDIM};
    const int gemmTiles = (NNODES / 16) * (HDIM / 16);
    const long long edgeWork = (long long)NEDGES * (HDIM / 4);
    const int edgeBlocks = (int)((edgeWork + TB - 1) / TB);

    for (int l = 0; l < 3; ++l) {
        wmma_gemm_f32<<<gemmTiles, 32, 0, stream>>>(hin, W[l], hw, nullptr,
                                                    NNODES, Kdim[l], HDIM, 0);
        float* agg = aggbuf[l];
        hipMemsetAsync(agg, 0, (size_t)NNODES * HDIM * sizeof(float), stream);
        edge_agg<<<edgeBlocks, TB, 0, stream>>>(hw, src, dst, dinv, agg);
        bn_relu_self<<<nfBlocks, TB, 0, stream>>>(hw, dinv, b[l],
                                                  bn_g + l * HDIM, bn_b + l * HDIM,
                                                  bn_m + l * HDIM, bn_v + l * HDIM,
                                                  agg);
        hin = agg;
    }

    // global mean pool
    hipMemsetAsync(pooled, 0, (size_t)NG * HDIM * sizeof(float), stream);
    hipMemsetAsync(counts, 0, (size_t)NG * sizeof(float), stream);
    count_kernel<<<(NNODES + TB - 1) / TB, TB, 0, stream>>>(batch, counts);
    pool_sum<<<nfBlocks, TB, 0, stream>>>(hin, batch, pooled);
    pool_div<<<(NG * HDIM + TB - 1) / TB, TB, 0, stream>>>(pooled, counts);

    // MLP head (bias+relu fused into WMMA epilogue)
    wmma_gemm_f32<<<(NG / 16) * (HDIM / 16), 32, 0, stream>>>(pooled, mlp_w1, z1,
                                                              mlp_b1, NG, HDIM, HDIM, 1);
    wmma_gemm_f32<<<(NG / 16) * (NL / 16), 32, 0, stream>>>(z1, mlp_w2, (float*)d_out,
                                                            mlp_b2, NG, HDIM, NL, 0);
}